// GATNet_19739669692885
// MI455X (gfx1250) — compile-verified
//
#include <hip/hip_runtime.h>
#include <hip/hip_bf16.h>
#include <math.h>

typedef __attribute__((ext_vector_type(2))) float v2f;
typedef __attribute__((ext_vector_type(8))) float v8f;

#define NEG_SLOPE 0.2f

// ---------- monotonic-uint encoding for float atomic max ----------
__device__ __forceinline__ unsigned f2key(float f) {
    unsigned b = __float_as_uint(f);
    return (b & 0x80000000u) ? ~b : (b | 0x80000000u);
}
__device__ __forceinline__ float key2f(unsigned k) {
    return __uint_as_float((k & 0x80000000u) ? (k ^ 0x80000000u) : ~k);
}
#define NEGINF_KEY 0x007FFFFFu  // f2key(-inf)

// ---------- init: seed max-keys with -inf, zero denominators/accumulators ----------
__global__ void gat_init(unsigned* __restrict__ mk1, unsigned* __restrict__ mk2,
                         float* __restrict__ dn1, float* __restrict__ dn2,
                         float* __restrict__ ag1, float* __restrict__ ag2, int N) {
    int n = blockIdx.x * blockDim.x + threadIdx.x;
    if (n >= N) return;
    mk1[n] = NEGINF_KEY; mk2[n] = NEGINF_KEY;
    dn1[n] = 0.f;        dn2[n] = 0.f;
#pragma unroll
    for (int f = 0; f < 8; ++f) ag1[n * 8 + f] = 0.f;
    ag2[n * 2 + 0] = 0.f;
    ag2[n * 2 + 1] = 0.f;
}

// ---------- fused linear + attention-logit kernel (WMMA path) ----------
// 256 threads = 8 waves; each wave32 computes one 16-row tile:
// H[16,NOUT] = X[16,K] @ W[K,NOUT] via V_WMMA_F32_16X16X4_F32 (K stepped by 4).
// A/B fragments are fully preloaded (branch-free, clamped addressing) so the
// MAC loop is a clean WMMA chain. The D tile is staged per-wave in LDS and
// lanes 0-15 compute alpha_src[row], lanes 16-31 compute alpha_dst[row].
template <int K, int NOUT>
__global__ __launch_bounds__(256) void lin_wmma(
        const float* __restrict__ X, const float* __restrict__ W,
        const float* __restrict__ a_s, const float* __restrict__ a_d,
        float* __restrict__ H, float* __restrict__ alpha_s,
        float* __restrict__ alpha_d, int N) {
    __shared__ float tile[8][16][17];

    const int wv   = threadIdx.x >> 5;     // wave in block: 0..7
    const int lane = threadIdx.x & 31;
    const int half = lane >> 4;            // 0 or 1
    const int l    = lane & 15;

    const int numTiles = (N + 15) >> 4;
    int t  = blockIdx.x * 8 + wv;
    int tc = t < numTiles ? t : (numTiles - 1);   // clamp: loads stay in range
    int row  = tc * 16 + l;
    int rclp = row < N ? row : (N - 1);           // EXEC stays uniform

    // ---- preload B fragments (W is tiny & uniform; branch-free via clamp+select)
    const int colc = (l < NOUT) ? l : 0;
    v2f bfrag[K / 4];
#pragma unroll
    for (int kk = 0; kk < K; kk += 4) {
        const int ka = kk + half * 2;
        const float b0 = W[ka * NOUT + colc];
        const float b1 = W[(ka + 1) * NOUT + colc];
        bfrag[kk / 4].x = (l < NOUT) ? b0 : 0.f;
        bfrag[kk / 4].y = (l < NOUT) ? b1 : 0.f;
    }
    // ---- preload A fragments (contiguous pair -> global_load_b64)
    v2f afrag[K / 4];
#pragma unroll
    for (int kk = 0; kk < K; kk += 4) {
        const int ka = kk + half * 2;
        afrag[kk / 4].x = X[(size_t)rclp * K + ka];
        afrag[kk / 4].y = X[(size_t)rclp * K + ka + 1];
    }

    // ---- MAC chain
    v8f c = {};
#pragma unroll
    for (int kk = 0; kk < K; kk += 4) {
        c = __builtin_amdgcn_wmma_f32_16x16x4_f32(
                /*neg_a=*/false, afrag[kk / 4], /*neg_b=*/false, bfrag[kk / 4],
                /*c_mod=*/(short)0, c, /*reuse_a=*/false, /*reuse_b=*/false);
    }

    // C/D 16x16 f32: VGPR v, lanes 0-15 -> M=v, lanes 16-31 -> M=v+8, N=l
    const bool live = (t < numTiles);
#pragma unroll
    for (int v = 0; v < 8; ++v) {
        const int M = v + half * 8;
        tile[wv][M][l] = c[v];
        const int orow = t * 16 + M;
        if (live && l < NOUT && orow < N) H[(size_t)orow * NOUT + l] = c[v];
    }
    __syncthreads();

    // lanes 0-15: alpha_src for rows 0..15; lanes 16-31: alpha_dst
    const float* avec = half ? a_d : a_s;
    float s = 0.f;
#pragma unroll
    for (int f = 0; f < NOUT; ++f) s += tile[wv][l][f] * avec[f];
    const int orow = t * 16 + l;
    if (live && orow < N) {
        if (half == 0) alpha_s[orow] = s;
        else           alpha_d[orow] = s;
    }
}

// ---------- edge pass 1: leaky-relu score + segment max ----------
__global__ void edge_scores(const int* __restrict__ src, const int* __restrict__ dst,
                            const float* __restrict__ as, const float* __restrict__ ad,
                            float* __restrict__ e_out, unsigned* __restrict__ mkey,
                            int E) {
    int i = blockIdx.x * blockDim.x + threadIdx.x;
    if (i >= E) return;
    const int s = src[i];
    const int d = dst[i];
    float e = as[s] + ad[d];
    e = (e > 0.f) ? e : NEG_SLOPE * e;       // leaky_relu
    e_out[i] = e;
    atomicMax(&mkey[d], f2key(e));
}

// ---------- edge pass 2: exp(e - max) + segment sum ----------
__global__ void edge_expsum(const int* __restrict__ dst, float* __restrict__ e_buf,
                            const unsigned* __restrict__ mkey,
                            float* __restrict__ denom, int E) {
    int i = blockIdx.x * blockDim.x + threadIdx.x;
    if (i >= E) return;
    const int d = dst[i];
    const float m = key2f(mkey[d]);
    const float ee = expf(e_buf[i] - m);
    e_buf[i] = ee;
    atomicAdd(&denom[d], ee);
}

// ---------- edge pass 3 (F=8): vectorized gather + normalized scatter ----------
__global__ void edge_aggregate8(const int* __restrict__ src, const int* __restrict__ dst,
                                const float* __restrict__ e_buf,
                                const float* __restrict__ denom,
                                const float* __restrict__ h, float* __restrict__ agg,
                                int E) {
    int i = blockIdx.x * blockDim.x + threadIdx.x;
    if (i >= E) return;
    const int s = src[i];
    const int d = dst[i];
    __builtin_prefetch(&h[(size_t)s * 8], 0, 0);   // global_prefetch_b8
    const float alpha = e_buf[i] / denom[d];
    const float4* h4 = (const float4*)(h + (size_t)s * 8);   // 32B row, b128 loads
    const float4 ha = h4[0];
    const float4 hb = h4[1];
    float* ap = agg + (size_t)d * 8;
    atomicAdd(ap + 0, alpha * ha.x);
    atomicAdd(ap + 1, alpha * ha.y);
    atomicAdd(ap + 2, alpha * ha.z);
    atomicAdd(ap + 3, alpha * ha.w);
    atomicAdd(ap + 4, alpha * hb.x);
    atomicAdd(ap + 5, alpha * hb.y);
    atomicAdd(ap + 6, alpha * hb.z);
    atomicAdd(ap + 7, alpha * hb.w);
}

// ---------- edge pass 3 (F=2) ----------
__global__ void edge_aggregate2(const int* __restrict__ src, const int* __restrict__ dst,
                                const float* __restrict__ e_buf,
                                const float* __restrict__ denom,
                                const float* __restrict__ h, float* __restrict__ agg,
                                int E) {
    int i = blockIdx.x * blockDim.x + threadIdx.x;
    if (i >= E) return;
    const int s = src[i];
    const int d = dst[i];
    const float alpha = e_buf[i] / denom[d];
    const float2 hv = *(const float2*)(h + (size_t)s * 2);   // b64 load
    float* ap = agg + (size_t)d * 2;
    atomicAdd(ap + 0, alpha * hv.x);
    atomicAdd(ap + 1, alpha * hv.y);
}

// ---------- node pass: h2 = relu(agg1 + b1) ----------
__global__ void relu_bias(const float* __restrict__ agg, const float* __restrict__ b,
                          float* __restrict__ out, int N) {
    int n = blockIdx.x * blockDim.x + threadIdx.x;
    if (n >= N) return;
#pragma unroll
    for (int f = 0; f < 8; ++f)
        out[n * 8 + f] = fmaxf(agg[n * 8 + f] + b[f], 0.f);
}

// ---------- final: out = log_softmax(agg2 + b2) over 2 cols ----------
__global__ void final_logsoftmax(const float* __restrict__ agg,
                                 const float* __restrict__ b,
                                 float* __restrict__ out, int N) {
    int n = blockIdx.x * blockDim.x + threadIdx.x;
    if (n >= N) return;
    const float z0 = agg[n * 2 + 0] + b[0];
    const float z1 = agg[n * 2 + 1] + b[1];
    const float m = fmaxf(z0, z1);
    const float lse = m + logf(expf(z0 - m) + expf(z1 - m));
    out[n * 2 + 0] = z0 - lse;
    out[n * 2 + 1] = z1 - lse;
}

extern "C" void kernel_launch(void* const* d_in, const int* in_sizes, int n_in,
                              void* d_out, int out_size, void* d_ws, size_t ws_size,
                              hipStream_t stream) {
    const float* x   = (const float*)d_in[0];
    const int*   ei  = (const int*)d_in[1];     // [2, E] int32
    const float* W1  = (const float*)d_in[2];   // [36, 8]
    const float* as1 = (const float*)d_in[3];
    const float* ad1 = (const float*)d_in[4];
    const float* b1  = (const float*)d_in[5];
    const float* W2  = (const float*)d_in[6];   // [8, 2]
    const float* as2 = (const float*)d_in[7];
    const float* ad2 = (const float*)d_in[8];
    const float* b2  = (const float*)d_in[9];
    float* out = (float*)d_out;

    const int N = in_sizes[0] / 36;
    const int E = in_sizes[1] / 2;
    const int* src = ei;
    const int* dst = ei + E;

    // workspace carve-up (floats; d_ws is >=256B aligned so float4 rows are fine)
    float* ws = (float*)d_ws;
    float*    h1  = ws;            ws += (size_t)N * 8;
    float*    s1  = ws;            ws += N;
    float*    t1  = ws;            ws += N;
    unsigned* mk1 = (unsigned*)ws; ws += N;
    float*    dn1 = ws;            ws += N;
    float*    ag1 = ws;            ws += (size_t)N * 8;
    float*    h2  = ws;            ws += (size_t)N * 8;
    float*    z2  = ws;            ws += (size_t)N * 2;
    float*    s2  = ws;            ws += N;
    float*    t2  = ws;            ws += N;
    unsigned* mk2 = (unsigned*)ws; ws += N;
    float*    dn2 = ws;            ws += N;
    float*    ag2 = ws;            ws += (size_t)N * 2;
    float*    eb  = ws;            ws += E;

    const int nb  = (N + 255) / 256;
    const int ebk = (E + 255) / 256;
    const int numTiles = (N + 15) / 16;
    const int lb  = (numTiles + 7) / 8;          // 8 waves (tiles) per block

    gat_init<<<nb, 256, 0, stream>>>(mk1, mk2, dn1, dn2, ag1, ag2, N);

    // ---- layer 1 ----
    lin_wmma<36, 8><<<lb, 256, 0, stream>>>(x, W1, as1, ad1, h1, s1, t1, N);
    edge_scores<<<ebk, 256, 0, stream>>>(src, dst, s1, t1, eb, mk1, E);
    edge_expsum<<<ebk, 256, 0, stream>>>(dst, eb, mk1, dn1, E);
    edge_aggregate8<<<ebk, 256, 0, stream>>>(src, dst, eb, dn1, h1, ag1, E);
    relu_bias<<<nb, 256, 0, stream>>>(ag1, b1, h2, N);

    // ---- layer 2 ----
    lin_wmma<8, 2><<<lb, 256, 0, stream>>>(h2, W2, as2, ad2, z2, s2, t2, N);
    edge_scores<<<ebk, 256, 0, stream>>>(src, dst, s2, t2, eb, mk2, E);
    edge_expsum<<<ebk, 256, 0, stream>>>(dst, eb, mk2, dn2, E);
    edge_aggregate2<<<ebk, 256, 0, stream>>>(src, dst, eb, dn2, z2, ag2, E);

    final_logsoftmax<<<nb, 256, 0, stream>>>(ag2, b2, out, N);
}